// FasterRCNNTrainer_87943750352915
// MI455X (gfx1250) — compile-verified
//
#include <hip/hip_runtime.h>
#include <hip/hip_bf16.h>
#include <stdint.h>

// Problem constants (match reference)
#define BB 16     // batch
#define TT 128    // targets
#define RR 8192   // proposals
#define CC 91     // classes
#define BLK 256
#define IOU_THR 0.7f
// Max tile-copy async ops issued per wave after the table ops (ceil(5824/256)=23)
#define TILE_OPS 23

// ---------------- CDNA5 async global->LDS staging ----------------
#if defined(__has_builtin)
#  if __has_builtin(__builtin_amdgcn_global_load_async_to_lds_b128)
#    define HAVE_ASYNC_LDS 1
#  endif
#  if __has_builtin(__builtin_amdgcn_s_wait_asynccnt)
#    define HAVE_WAIT_ASYNC 1
#  endif
#endif
#ifndef HAVE_ASYNC_LDS
#  define HAVE_ASYNC_LDS 0
#endif
#ifndef HAVE_WAIT_ASYNC
#  define HAVE_WAIT_ASYNC 0
#endif

#if HAVE_ASYNC_LDS
#  if HAVE_WAIT_ASYNC
#    define WAIT_ASYNC(n) __builtin_amdgcn_s_wait_asynccnt(n)
#  else
#    define WAIT_ASYNC(n) asm volatile("s_wait_asynccnt %0" ::"i"(n) : "memory")
#  endif
#else
#  define WAIT_ASYNC(n) ((void)0)
#endif

// Builtin signature (from clang diagnostic): arg0 = int4 in AS1 (global),
// arg1 = int4 in AS3 (LDS), arg2 = imm offset, arg3 = cpol.
typedef int v4i __attribute__((vector_size(16)));
typedef __attribute__((address_space(1))) v4i* gv4i_p;
typedef __attribute__((address_space(3))) v4i* lv4i_p;

// Copy 16 bytes global -> LDS: global_load_async_to_lds_b128 (ASYNCcnt path).
__device__ __forceinline__ void cp16_g2l(void* lds_dst, const void* gsrc) {
#if HAVE_ASYNC_LDS
  __builtin_amdgcn_global_load_async_to_lds_b128(
      (gv4i_p)(uintptr_t)gsrc,
      (lv4i_p)(uint32_t)(uintptr_t)lds_dst,
      0, 0);
#else
  *reinterpret_cast<float4*>(lds_dst) = *reinterpret_cast<const float4*>(gsrc);
#endif
}

__device__ __forceinline__ float smooth_l1(float x) {
  float ax = fabsf(x);
  return (ax < 1.0f) ? 0.5f * x * x : ax - 0.5f;
}

// ---------------- main kernel: one block = one batch-row tile of 256 proposals ----------------
__global__ void __launch_bounds__(BLK)
rcnn_loss_main(const float* __restrict__ nms_reg,    // [B,R,4]
               const float* __restrict__ rcnn_reg,   // [B,R,4]
               const float* __restrict__ rcnn_cls,   // [B,R,C]
               const float* __restrict__ bboxes,     // [B,T,4]
               const int*   __restrict__ classes,    // [B,T]
               const int*   __restrict__ reduction,  // scalar
               float* __restrict__ acc)              // [4]: count, nll, correct, reg
{
  __shared__ __align__(16) float s_bb[TT * 4];        // 2 KB
  __shared__ __align__(16) int   s_cl[TT];            // 0.5 KB
  __shared__ __align__(16) float s_tile[BLK * CC];    // 93,184 B (fits 320KB WGP LDS)
  __shared__ float s_red[BLK / 32][4];

  const int b   = blockIdx.y;
  const int r0  = blockIdx.x * BLK;
  const int tid = threadIdx.x;
  const int r   = r0 + tid;

  // ---- phase 1: issue table async loads FIRST (in-order completion matters) ----
  if (tid < 128) {
    cp16_g2l(&s_bb[tid * 4], bboxes + (size_t)b * TT * 4 + tid * 4);
  } else if (tid < 160) {
    const int i = tid - 128;
    cp16_g2l(&s_cl[i * 4], classes + b * TT + i * 4);
  }
  // ---- phase 2: issue the 93KB class-logit tile (<= TILE_OPS b128 per lane) ----
  const float* gtile = rcnn_cls + ((size_t)b * RR + r0) * CC;  // contiguous, 16B-aligned
  for (int i = tid; i < (BLK * CC) / 4; i += BLK)
    cp16_g2l(&s_tile[i * 4], gtile + i * 4);

  // Per-thread box loads (LOADcnt) also overlap the async tile stream.
  const float4 a  = *reinterpret_cast<const float4*>(nms_reg  + ((size_t)b * RR + r) * 4);
  const float4 rg = *reinterpret_cast<const float4*>(rcnn_reg + ((size_t)b * RR + r) * 4);

  // Async loads complete in order per wave: ASYNCcnt <= TILE_OPS guarantees the
  // earlier table ops are done while the big tile keeps streaming.
  WAIT_ASYNC(TILE_OPS);
  __syncthreads();

  // ---- IoU argmax over T targets (LDS broadcast reads) — overlaps tile DMA ----
  const float area_a = (a.z - a.x) * (a.w - a.y);
  float best = -1.0f;
  int   bi   = 0;
#pragma unroll 4
  for (int t = 0; t < TT; ++t) {
    const float bt = s_bb[t * 4 + 0], bl = s_bb[t * 4 + 1];
    const float bm = s_bb[t * 4 + 2], br = s_bb[t * 4 + 3];
    const float it = fmaxf(a.x, bt), il = fmaxf(a.y, bl);
    const float ib = fminf(a.z, bm), ir = fminf(a.w, br);
    const float inter  = fmaxf(ib - it, 0.0f) * fmaxf(ir - il, 0.0f);
    const float area_b = (bm - bt) * (br - bl);
    const float iou    = inter / (area_a + area_b - inter);
    if (iou > best) { best = iou; bi = t; }   // first-max, matches jnp.argmax
  }
  const float mask = (best > IOU_THR) ? 1.0f : 0.0f;

  // ---- smooth-L1 regression terms on best box ----
  const float bt = s_bb[bi * 4 + 0], bl = s_bb[bi * 4 + 1];
  const float bm = s_bb[bi * 4 + 2], br = s_bb[bi * 4 + 3];
  const float red = (float)reduction[0];
  const float ro0 = rintf(bt / red) * red;   // rint = round-half-even, matches jnp.round
  const float ro1 = rintf(bl / red) * red;
  const float ro2 = rintf(bm / red) * red;
  const float ro3 = rintf(br / red) * red;
  float hg = ro2 - ro0;  hg = (hg == 0.0f) ? 1.0f : hg;
  float wd = ro3 - ro1;  wd = (wd == 0.0f) ? 1.0f : wd;
  const float regl = smooth_l1(rg.x - (bt - ro0) / hg)
                   + smooth_l1(rg.y - (bl - ro1) / wd)
                   + smooth_l1(rg.z - (bm - ro2) / hg)
                   + smooth_l1(rg.w - (br - ro3) / wd);

  // ---- now require the class-logit tile ----
  WAIT_ASYNC(0);
  __syncthreads();

  // ---- log-softmax NLL + accuracy over C classes (LDS row, stride 91 dwords:
  //      27 mod 64 banks, coprime with 64 -> conflict-free) ----
  const float* row = &s_tile[tid * CC];
  float m = row[0];
  int   am = 0;
  for (int c = 1; c < CC; ++c) {
    const float x = row[c];
    if (x > m) { m = x; am = c; }
  }
  float s = 0.0f;
  for (int c = 0; c < CC; ++c) s += __expf(row[c] - m);
  const int   tgt     = s_cl[bi];
  const float nll     = m + __logf(s) - row[tgt];
  const float correct = (am == tgt) ? 1.0f : 0.0f;

  // ---- block reduction: wave32 shuffle tree, then cross-wave via LDS ----
  float v0 = mask, v1 = nll * mask, v2 = correct * mask, v3 = regl * mask;
#pragma unroll
  for (int off = 16; off > 0; off >>= 1) {
    v0 += __shfl_down(v0, off, 32);
    v1 += __shfl_down(v1, off, 32);
    v2 += __shfl_down(v2, off, 32);
    v3 += __shfl_down(v3, off, 32);
  }
  const int lane = tid & 31, wv = tid >> 5;
  if (lane == 0) { s_red[wv][0] = v0; s_red[wv][1] = v1; s_red[wv][2] = v2; s_red[wv][3] = v3; }
  __syncthreads();
  if (tid == 0) {
    float t0 = 0.f, t1 = 0.f, t2 = 0.f, t3 = 0.f;
#pragma unroll
    for (int w = 0; w < BLK / 32; ++w) {
      t0 += s_red[w][0]; t1 += s_red[w][1]; t2 += s_red[w][2]; t3 += s_red[w][3];
    }
    atomicAdd(&acc[0], t0);
    atomicAdd(&acc[1], t1);
    atomicAdd(&acc[2], t2);
    atomicAdd(&acc[3], t3);
  }
}

// ---------------- finalize: three masked means ----------------
__global__ void rcnn_loss_finalize(const float* __restrict__ acc, float* __restrict__ out) {
  if (threadIdx.x == 0 && blockIdx.x == 0) {
    const float cnt = acc[0];
    const float den = fmaxf(cnt, 1.0f);
    const float pos = (cnt > 0.0f) ? 1.0f : 0.0f;
    out[0] = (acc[1] / den) * pos;  // cls_loss
    out[1] = (acc[3] / den) * pos;  // reg_loss
    out[2] = (acc[2] / den) * pos;  // accuracy
  }
}

extern "C" void kernel_launch(void* const* d_in, const int* in_sizes, int n_in,
                              void* d_out, int out_size, void* d_ws, size_t ws_size,
                              hipStream_t stream) {
  const float* nms_reg   = (const float*)d_in[0];
  // d_in[1] = nms_cls, unused by the loss
  const float* rcnn_reg  = (const float*)d_in[2];
  const float* rcnn_cls  = (const float*)d_in[3];
  const float* bboxes    = (const float*)d_in[4];
  const int*   classes   = (const int*)d_in[5];
  const int*   reduction = (const int*)d_in[6];
  float* out = (float*)d_out;
  float* acc = (float*)d_ws;  // 4 float accumulators

  (void)hipMemsetAsync(acc, 0, 4 * sizeof(float), stream);

  dim3 grid(RR / BLK, BB);  // 32 x 16 blocks
  rcnn_loss_main<<<grid, BLK, 0, stream>>>(nms_reg, rcnn_reg, rcnn_cls,
                                           bboxes, classes, reduction, acc);
  rcnn_loss_finalize<<<1, 32, 0, stream>>>(acc, out);
}